// ADUnit_43671227466564
// MI455X (gfx1250) — compile-verified
//
#include <hip/hip_runtime.h>
#include <math.h>

typedef float v2f __attribute__((ext_vector_type(2)));
typedef float v8f __attribute__((ext_vector_type(8)));

#define N_SAMPLES 16384
#define L_DIM     4096
#define D_DIM     1024
#define SHRINK_T  0.0025f
#define EPS_V     1e-12f

#define WMMA_F32(a, b, c) \
  __builtin_amdgcn_wmma_f32_16x16x4_f32(false, (a), false, (b), (short)0, (c), false, false)

// ---------------------------------------------------------------------------
// Top-2 (argmax + second argmax) helpers, jnp.argmax tie-break = lowest index.
// ---------------------------------------------------------------------------
struct Top2 { float v1; int i1; float v2; int i2; };

__device__ __forceinline__ bool better(float a, int ia, float b, int ib) {
  return (a > b) || (a == b && ia < ib);
}

__device__ __forceinline__ Top2 t2_merge(Top2 a, Top2 b) {
  Top2 r;
  if (better(a.v1, a.i1, b.v1, b.i1)) {
    r.v1 = a.v1; r.i1 = a.i1;
    if (better(a.v2, a.i2, b.v1, b.i1)) { r.v2 = a.v2; r.i2 = a.i2; }
    else                                { r.v2 = b.v1; r.i2 = b.i1; }
  } else {
    r.v1 = b.v1; r.i1 = b.i1;
    if (better(b.v2, b.i2, a.v1, a.i1)) { r.v2 = b.v2; r.i2 = b.i2; }
    else                                { r.v2 = a.v1; r.i2 = a.i1; }
  }
  return r;
}

// ---------------------------------------------------------------------------
// Fused: GEMM1 (WMMA f32, S tile -> 256 KB LDS) + softmax/shrink/L1 (in LDS)
// + att writeback + nl/nl2 gather + GEMM2 (A from LDS, B from L2-resident W).
// One workgroup (256 thr = 8 wave32) per 16-sample tile.
// 4-way register blocking: 4 independent accumulator chains per wave hide the
// dependent-WMMA pipeline latency and cut VMEM ops per WMMA.
// ---------------------------------------------------------------------------
__global__ __launch_bounds__(256) void fused_memnet(const float* __restrict__ x,
                                                    const float* __restrict__ w,
                                                    float* __restrict__ out,
                                                    float* __restrict__ att,
                                                    float* __restrict__ nl,
                                                    float* __restrict__ nl2) {
  __shared__ float s_att[16 * L_DIM];      // 256 KB logits / att tile
  __shared__ float red[16][17];            // per-row partials (+1 result slot)
  __shared__ Top2  t2s[256];
  __shared__ int   s_ind[16], s_ind2[16];

  const int tid  = threadIdx.x;
  const int lane = tid & 31;
  const int wid  = tid >> 5;
  const int half = lane >> 4;              // WMMA f32 A/B: K pair select
  const int m16  = lane & 15;
  const int row_base = blockIdx.x * 16;

  // ------------------- Phase 1: S = x_tile @ W^T  (into LDS) ---------------
  {
    const float* arow = x + (size_t)(row_base + m16) * D_DIM + 2 * half;
    for (int g = 0; g < 8; ++g) {          // 8 groups x 4 tiles x 16 = 512 cols
      const int col0 = wid * 512 + g * 64;
      const float* b0 = w + (size_t)(col0 +  0 + m16) * D_DIM + 2 * half;
      const float* b1 = w + (size_t)(col0 + 16 + m16) * D_DIM + 2 * half;
      const float* b2 = w + (size_t)(col0 + 32 + m16) * D_DIM + 2 * half;
      const float* b3 = w + (size_t)(col0 + 48 + m16) * D_DIM + 2 * half;
      v8f acc0 = {}, acc1 = {}, acc2 = {}, acc3 = {};
#pragma unroll 2
      for (int k = 0; k < D_DIM; k += 4) {
        v2f a   = *(const v2f*)(arow + k);  // A fragment shared by 4 WMMAs
        v2f vb0 = *(const v2f*)(b0 + k);
        v2f vb1 = *(const v2f*)(b1 + k);
        v2f vb2 = *(const v2f*)(b2 + k);
        v2f vb3 = *(const v2f*)(b3 + k);
        acc0 = WMMA_F32(a, vb0, acc0);
        acc1 = WMMA_F32(a, vb1, acc1);
        acc2 = WMMA_F32(a, vb2, acc2);
        acc3 = WMMA_F32(a, vb3, acc3);
      }
#pragma unroll
      for (int r8 = 0; r8 < 8; ++r8) {
        const int base = (r8 + 8 * half) * L_DIM + m16 + col0;
        s_att[base     ] = acc0[r8];
        s_att[base + 16] = acc1[r8];
        s_att[base + 32] = acc2[r8];
        s_att[base + 48] = acc3[r8];
      }
    }
  }
  __syncthreads();

  // ------------------- Phase 2: softmax + shrink + L1 + top-2 (in LDS) -----
  const int r = tid >> 4;                  // row 0..15
  const int c = tid & 15;                  // 16 threads cooperate per row
  float* srow = s_att + r * L_DIM;

  // row max
  float m = -INFINITY;
  for (int i = c; i < L_DIM; i += 16) m = fmaxf(m, srow[i]);
  red[r][c] = m;
  __syncthreads();
  if (c == 0) {
    float mm = red[r][0];
#pragma unroll
    for (int j = 1; j < 16; ++j) mm = fmaxf(mm, red[r][j]);
    red[r][16] = mm;
  }
  __syncthreads();
  const float rowmax = red[r][16];
  __syncthreads();

  // exp + sum
  float sum = 0.f;
  for (int i = c; i < L_DIM; i += 16) {
    float e = expf(srow[i] - rowmax);
    srow[i] = e;
    sum += e;
  }
  red[r][c] = sum;
  __syncthreads();
  if (c == 0) {
    float ss = red[r][0];
#pragma unroll
    for (int j = 1; j < 16; ++j) ss += red[r][j];
    red[r][16] = ss;
  }
  __syncthreads();
  const float inv = 1.f / red[r][16];
  __syncthreads();

  // hard_shrink_relu + L1 accumulate (values >= 0)
  float l1 = 0.f;
  for (int i = c; i < L_DIM; i += 16) {
    float a  = srow[i] * inv;
    float sh = a - SHRINK_T;
    float v  = ((sh > 0.f ? sh : 0.f) * a) / (fabsf(sh) + EPS_V);
    srow[i] = v;
    l1 += v;
  }
  red[r][c] = l1;
  __syncthreads();
  if (c == 0) {
    float ss = red[r][0];
#pragma unroll
    for (int j = 1; j < 16; ++j) ss += red[r][j];
    red[r][16] = ss;
  }
  __syncthreads();
  const float invl1 = 1.f / fmaxf(red[r][16], EPS_V);
  __syncthreads();

  // normalize in place + local top-2
  Top2 t; t.v1 = -INFINITY; t.i1 = 0x7FFFFFFF; t.v2 = -INFINITY; t.i2 = 0x7FFFFFFF;
  for (int i = c; i < L_DIM; i += 16) {
    float a = srow[i] * invl1;
    srow[i] = a;
    if (better(a, i, t.v1, t.i1))      { t.v2 = t.v1; t.i2 = t.i1; t.v1 = a; t.i1 = i; }
    else if (better(a, i, t.v2, t.i2)) { t.v2 = a; t.i2 = i; }
  }
  t2s[tid] = t;
  __syncthreads();
  if (c == 0) {
    Top2 acc = t2s[r * 16];
#pragma unroll
    for (int j = 1; j < 16; ++j) acc = t2_merge(acc, t2s[r * 16 + j]);
    s_ind[r]  = acc.i1;
    s_ind2[r] = acc.i2;
  }
  __syncthreads();   // s_att fully normalized + indices published past here

  // ------------------- Phase 3: writebacks (att, nl, nl2) -------------------
  {
    float* abase = att + (size_t)row_base * L_DIM;
    for (int idx = tid; idx < 16 * L_DIM; idx += 256) {
      abase[idx] = s_att[idx];            // coalesced row-major tile copy
    }
  }
  {
    for (int idx = tid; idx < 16 * D_DIM; idx += 256) {
      const int rr = idx >> 10;           // D_DIM = 1024
      const int dd = idx & (D_DIM - 1);
      nl [(size_t)(row_base + rr) * D_DIM + dd] = w[(size_t)s_ind [rr] * D_DIM + dd];
      nl2[(size_t)(row_base + rr) * D_DIM + dd] = w[(size_t)s_ind2[rr] * D_DIM + dd];
    }
  }
  // no barrier needed: phase 4 only reads s_att, final since the last barrier

  // ------------------- Phase 4: out = att_tile(LDS) @ W ---------------------
  {
    const float* alds = s_att + m16 * L_DIM + 2 * half;   // A fragment (LDS)
    for (int g = 0; g < 2; ++g) {         // 2 groups x 4 tiles x 16 = 128 cols
      const int d0 = wid * 128 + g * 64 + m16;
      const float* bc0 = w + (size_t)(2 * half) * D_DIM + d0;
      const float* bc1 = bc0 + 16;
      const float* bc2 = bc0 + 32;
      const float* bc3 = bc0 + 48;
      v8f acc0 = {}, acc1 = {}, acc2 = {}, acc3 = {};
#pragma unroll 2
      for (int l = 0; l < L_DIM; l += 4) {
        v2f a = *(const v2f*)(alds + l);                  // ds_load_b64, shared
        const size_t o = (size_t)l * D_DIM;
        v2f vb0, vb1, vb2, vb3;
        vb0.x = bc0[o]; vb0.y = bc0[o + D_DIM];
        vb1.x = bc1[o]; vb1.y = bc1[o + D_DIM];
        vb2.x = bc2[o]; vb2.y = bc2[o + D_DIM];
        vb3.x = bc3[o]; vb3.y = bc3[o + D_DIM];
        acc0 = WMMA_F32(a, vb0, acc0);
        acc1 = WMMA_F32(a, vb1, acc1);
        acc2 = WMMA_F32(a, vb2, acc2);
        acc3 = WMMA_F32(a, vb3, acc3);
      }
#pragma unroll
      for (int rr = 0; rr < 8; ++rr) {
        const size_t obase = (size_t)(row_base + rr + 8 * half) * D_DIM + d0;
        out[obase     ] = acc0[rr];
        out[obase + 16] = acc1[rr];
        out[obase + 32] = acc2[rr];
        out[obase + 48] = acc3[rr];
      }
    }
  }
}

// ---------------------------------------------------------------------------
// centroid of weight rows: cen[d] = mean_l W[l][d]
// ---------------------------------------------------------------------------
__global__ __launch_bounds__(256) void centroid_kernel(const float* __restrict__ w,
                                                       float* __restrict__ cen) {
  const int d = blockIdx.x * 256 + threadIdx.x;
  float s = 0.f;
  for (int l = 0; l < L_DIM; ++l) s += w[(size_t)l * D_DIM + d];
  cen[d] = s * (1.0f / (float)L_DIM);
}

// ---------------------------------------------------------------------------
// col_mask[n] = ||x[n] - cen|| < 1.0
// ---------------------------------------------------------------------------
__global__ __launch_bounds__(256) void mask_kernel(const float* __restrict__ x,
                                                   const float* __restrict__ cen,
                                                   float* __restrict__ mask) {
  __shared__ float red[256];
  const int row = blockIdx.x;
  const int tid = threadIdx.x;
  const float* xr = x + (size_t)row * D_DIM;
  float s = 0.f;
  for (int d = tid; d < D_DIM; d += 256) {
    float diff = xr[d] - cen[d];
    s += diff * diff;
  }
  red[tid] = s; __syncthreads();
  for (int st = 128; st > 0; st >>= 1) {
    if (tid < st) red[tid] += red[tid + st];
    __syncthreads();
  }
  if (tid == 0) mask[row] = (sqrtf(red[0]) < 1.0f) ? 1.0f : 0.0f;
}

// ---------------------------------------------------------------------------
extern "C" void kernel_launch(void* const* d_in, const int* in_sizes, int n_in,
                              void* d_out, int out_size, void* d_ws, size_t ws_size,
                              hipStream_t stream) {
  const float* x = (const float*)d_in[0];   // [N, D]
  const float* w = (const float*)d_in[1];   // [L, D]

  float* out  = (float*)d_out;                              // [N, D]
  float* att  = out  + (size_t)N_SAMPLES * D_DIM;           // [N, L]
  float* nl   = att  + (size_t)N_SAMPLES * L_DIM;           // [N, D]
  float* nl2  = nl   + (size_t)N_SAMPLES * D_DIM;           // [N, D]
  float* mask = nl2  + (size_t)N_SAMPLES * D_DIM;           // [N]

  float* cen = (float*)d_ws;                                // [D]

  fused_memnet<<<N_SAMPLES / 16, 256, 0, stream>>>(x, w, out, att, nl, nl2);
  centroid_kernel<<<D_DIM / 256, 256, 0, stream>>>(w, cen);
  mask_kernel<<<N_SAMPLES, 256, 0, stream>>>(x, cen, mask);
}